// VADER_79121887527171
// MI455X (gfx1250) — compile-verified
//
#include <hip/hip_runtime.h>

// Problem constants (match reference)
#define BB 8
#define LL 96
#define DD 512
#define RR 64
#define NT 47
#define NL 3

typedef float v2f __attribute__((ext_vector_type(2)));
typedef float v8f __attribute__((ext_vector_type(8)));

// ---------------------------------------------------------------------------
// Gram matrix of dep_emb_table rows: G[t1*48+t2] = dot(table[t1], table[t2]).
// Layer-invariant; replaces the (B,L,L,D) gather tensor entirely.
// ---------------------------------------------------------------------------
__global__ void k_gram(const float* __restrict__ table, float* __restrict__ G) {
    int id = blockIdx.x * blockDim.x + threadIdx.x;
    if (id >= 48 * 48) return;
    int t1 = id / 48, t2 = id % 48;
    float acc = 0.0f;
    if (t1 < NT && t2 < NT) {
        const float* a = table + (size_t)t1 * DD;
        const float* b = table + (size_t)t2 * DD;
        for (int d = 0; d < DD; ++d) acc += a[d] * b[d];
    }
    G[id] = acc;
}

// ---------------------------------------------------------------------------
// seq0 = text * (gamma / sqrt(1+eps)) + beta
// ---------------------------------------------------------------------------
__global__ void k_bn(const float* __restrict__ text, const float* __restrict__ gamma,
                     const float* __restrict__ beta, float* __restrict__ seq) {
    int idx = blockIdx.x * blockDim.x + threadIdx.x;
    int d = idx & (DD - 1);
    float inv = 1.0f / sqrtf(1.0f + 0.001f);
    seq[idx] = text[idx] * (gamma[d] * inv) + beta[d];
}

// ---------------------------------------------------------------------------
// Scores: S[b,i,j] = (seq[b,i]·seq[b,j] + mask*G[v1][v2]) / sqrt(D)
// One wave per 16x16 tile; 8*6*6 = 288 waves. K=512 in steps of 4 (f32 WMMA).
// ---------------------------------------------------------------------------
__global__ void k_score(const float* __restrict__ seq, const float* __restrict__ G,
                        const int* __restrict__ depv, float* __restrict__ S) {
    int wave = blockIdx.x * (blockDim.x >> 5) + (threadIdx.x >> 5);
    int lane = threadIdx.x & 31;
    int half = lane >> 4;
    int l15  = lane & 15;
    int b   = wave / 36;
    int rem = wave % 36;
    int ti = rem / 6, tj = rem % 6;
    const float* A  = seq + (size_t)b * LL * DD + (size_t)ti * 16 * DD;
    const float* Bt = seq + (size_t)b * LL * DD + (size_t)tj * 16 * DD; // B = A-style rows (X·X^T)
    v8f c = {};
    for (int kk = 0; kk < DD; kk += 4) {
        int k0 = kk + 2 * half;
        v2f a, bf;
        a.x  = A[(size_t)l15 * DD + k0];
        a.y  = A[(size_t)l15 * DD + k0 + 1];
        bf.x = Bt[(size_t)l15 * DD + k0];
        bf.y = Bt[(size_t)l15 * DD + k0 + 1];
        c = __builtin_amdgcn_wmma_f32_16x16x4_f32(false, a, false, bf, (short)0, c,
                                                  false, false);
    }
    const float scale = 0.04419417382415922f; // 1/sqrt(512)
    int j = tj * 16 + l15;
    for (int r = 0; r < 8; ++r) {
        int i = ti * 16 + r + 8 * half;
        int v1 = depv[((size_t)b * LL + i) * LL + j];
        int v2 = depv[((size_t)b * LL + j) * LL + i];
        float g = (v1 > 0 && v2 > 0) ? G[v1 * 48 + v2] : 0.0f;
        S[((size_t)b * LL + i) * LL + j] = (c[r] + g) * scale;
    }
}

// ---------------------------------------------------------------------------
// Softmax over full row, multiply by dep_adj -> attn; then bucket attention
// weights by type t = dep_value[b,j,i] into wtyp[b,i,t] (deterministic, no
// atomics). One block per (b,i) row, 128 threads.
// ---------------------------------------------------------------------------
__global__ void k_attn(const float* __restrict__ S, const float* __restrict__ adj,
                       const int* __restrict__ depv, float* __restrict__ attn,
                       float* __restrict__ wtyp) {
    int b = blockIdx.x / LL;
    int i = blockIdx.x % LL;
    int tid = threadIdx.x;
    __shared__ float red[128];
    __shared__ float arow[LL];
    __shared__ int   trow[LL];
    size_t rowoff = ((size_t)b * LL + i) * LL;

    float s = (tid < LL) ? S[rowoff + tid] : -1e30f;
    red[tid] = s;
    __syncthreads();
    for (int off = 64; off > 0; off >>= 1) {
        if (tid < off) red[tid] = fmaxf(red[tid], red[tid + off]);
        __syncthreads();
    }
    float mx = red[0];
    __syncthreads();
    float e = (tid < LL) ? expf(s - mx) : 0.0f;
    red[tid] = e;
    __syncthreads();
    for (int off = 64; off > 0; off >>= 1) {
        if (tid < off) red[tid] += red[tid + off];
        __syncthreads();
    }
    float sum = red[0];
    __syncthreads();

    float a = 0.0f;
    if (tid < LL) {
        a = (e / sum) * adj[rowoff + tid];
        attn[rowoff + tid] = a;
        arow[tid] = a;
        trow[tid] = depv[((size_t)b * LL + tid) * LL + i]; // dep_value[b, j, i]
    }
    __syncthreads();
    if (tid < 64) {
        float w = 0.0f;
        if (tid >= 1 && tid < NT) {
            for (int j = 0; j < LL; ++j)
                if (trow[j] == tid) w += arow[j];
        }
        wtyp[((size_t)b * LL + i) * 64 + tid] = w;
    }
}

// ---------------------------------------------------------------------------
// agg[b,i,:] = attn[b,i,:] @ seq[b,:,:]  (K=96)  +  wtyp[b,i,:] @ table (K=48)
// One wave per 16x16 tile of (L x D) per batch: 8*6*32 = 1536 waves.
// ---------------------------------------------------------------------------
__global__ void k_agg(const float* __restrict__ attn, const float* __restrict__ seq,
                      const float* __restrict__ wtyp, const float* __restrict__ table,
                      float* __restrict__ agg) {
    int wave = blockIdx.x * (blockDim.x >> 5) + (threadIdx.x >> 5);
    int lane = threadIdx.x & 31;
    int half = lane >> 4, l15 = lane & 15;
    int b   = wave / (6 * 32);
    int rem = wave % (6 * 32);
    int ti = rem / 32, tj = rem % 32;
    const float* A  = attn + (size_t)b * LL * LL + (size_t)ti * 16 * LL;
    const float* Bs = seq  + (size_t)b * LL * DD + (size_t)tj * 16;
    v8f c = {};
    for (int kk = 0; kk < LL; kk += 4) {
        int k0 = kk + 2 * half;
        v2f a, bf;
        a.x  = A[(size_t)l15 * LL + k0];
        a.y  = A[(size_t)l15 * LL + k0 + 1];
        bf.x = Bs[(size_t)k0 * DD + l15];
        bf.y = Bs[(size_t)(k0 + 1) * DD + l15];
        c = __builtin_amdgcn_wmma_f32_16x16x4_f32(false, a, false, bf, (short)0, c,
                                                  false, false);
    }
    const float* A2 = wtyp + ((size_t)b * LL + (size_t)ti * 16) * 64;
    for (int kk = 0; kk < 48; kk += 4) {
        int k0 = kk + 2 * half;
        v2f a, bf;
        a.x  = A2[(size_t)l15 * 64 + k0];
        a.y  = A2[(size_t)l15 * 64 + k0 + 1];
        bf.x = (k0 < NT)     ? table[(size_t)k0 * DD + tj * 16 + l15]       : 0.0f;
        bf.y = (k0 + 1 < NT) ? table[(size_t)(k0 + 1) * DD + tj * 16 + l15] : 0.0f;
        c = __builtin_amdgcn_wmma_f32_16x16x4_f32(false, a, false, bf, (short)0, c,
                                                  false, false);
    }
    int col = tj * 16 + l15;
    for (int r = 0; r < 8; ++r) {
        int row = ti * 16 + r + 8 * half;
        agg[((size_t)b * LL + row) * DD + col] = c[r];
    }
}

// ---------------------------------------------------------------------------
// seq_out = relu(agg @ W_l + b_l): (768x512)@(512x512). 48*32 = 1536 waves.
// ---------------------------------------------------------------------------
__global__ void k_fc(const float* __restrict__ agg, const float* __restrict__ W,
                     const float* __restrict__ bias, float* __restrict__ out) {
    int wave = blockIdx.x * (blockDim.x >> 5) + (threadIdx.x >> 5);
    int lane = threadIdx.x & 31;
    int half = lane >> 4, l15 = lane & 15;
    int mi = wave / 32, nj = wave % 32;
    const float* A  = agg + (size_t)mi * 16 * DD;
    const float* Bw = W + (size_t)nj * 16;
    v8f c = {};
    for (int kk = 0; kk < DD; kk += 4) {
        int k0 = kk + 2 * half;
        v2f a, bf;
        a.x  = A[(size_t)l15 * DD + k0];
        a.y  = A[(size_t)l15 * DD + k0 + 1];
        bf.x = Bw[(size_t)k0 * DD + l15];
        bf.y = Bw[(size_t)(k0 + 1) * DD + l15];
        c = __builtin_amdgcn_wmma_f32_16x16x4_f32(false, a, false, bf, (short)0, c,
                                                  false, false);
    }
    int col = nj * 16 + l15;
    float bv = bias[col];
    for (int r = 0; r < 8; ++r) {
        int row = mi * 16 + r + 8 * half;
        float v = c[r] + bv;
        out[(size_t)row * DD + col] = v > 0.0f ? v : 0.0f;
    }
}

// ---------------------------------------------------------------------------
// Masked average pool over L per (b, d).
// ---------------------------------------------------------------------------
__global__ void k_pool(const float* __restrict__ seq, const float* __restrict__ mask,
                       float* __restrict__ pools) {
    int b = blockIdx.x;
    int d = threadIdx.x;
    float sum = 0.0f, cnt = 0.0f;
    for (int i = 0; i < LL; ++i) {
        float m = mask[b * LL + i];
        sum += seq[((size_t)b * LL + i) * DD + d] * m;
        if (m != 0.0f) cnt += 1.0f;
    }
    pools[(size_t)b * DD + d] = sum / (cnt + 1e-10f);
}

// ---------------------------------------------------------------------------
// Ensemble (softmax over 3 layer weights) + final FC: (8x512)@(512x64)+b.
// ---------------------------------------------------------------------------
__global__ void k_out(const float* __restrict__ pools, const float* __restrict__ ens_lin,
                      const float* __restrict__ fcW, const float* __restrict__ fcb,
                      float* __restrict__ out) {
    int tid = threadIdx.x;
    float e0 = ens_lin[0], e1 = ens_lin[1], e2 = ens_lin[2];
    float mx = fmaxf(e0, fmaxf(e1, e2));
    float x0 = expf(e0 - mx), x1 = expf(e1 - mx), x2 = expf(e2 - mx);
    float inv = 1.0f / (x0 + x1 + x2);
    float w0 = x0 * inv, w1 = x1 * inv, w2 = x2 * inv;
    int b = tid >> 6, r = tid & 63;
    const float* p0 = pools + (size_t)b * DD;
    const float* p1 = pools + (size_t)(BB + b) * DD;
    const float* p2 = pools + (size_t)(2 * BB + b) * DD;
    float acc = fcb[r];
    for (int d = 0; d < DD; ++d) {
        float e = p0[d] * w0 + p1[d] * w1 + p2[d] * w2;
        acc += e * fcW[(size_t)d * RR + r];
    }
    out[tid] = acc;
}

// ---------------------------------------------------------------------------
extern "C" void kernel_launch(void* const* d_in, const int* in_sizes, int n_in,
                              void* d_out, int out_size, void* d_ws, size_t ws_size,
                              hipStream_t stream) {
    const float* text   = (const float*)d_in[0];
    const float* mask   = (const float*)d_in[1];
    const float* adj    = (const float*)d_in[2];
    const float* table  = (const float*)d_in[3];
    const float* gamma  = (const float*)d_in[4];
    const float* beta   = (const float*)d_in[5];
    const float* Wst    = (const float*)d_in[6];
    const float* bst    = (const float*)d_in[7];
    const float* enslin = (const float*)d_in[8];
    const float* fcW    = (const float*)d_in[9];
    const float* fcb    = (const float*)d_in[10];
    const int*   depv   = (const int*)d_in[11];

    float* ws   = (float*)d_ws;
    float* G    = ws;                       // 48*48
    float* seq0 = G + 48 * 48;              // B*L*D
    float* seq1 = seq0 + BB * LL * DD;      // B*L*D
    float* attn = seq1 + BB * LL * DD;      // B*L*L (scores, then attn in place)
    float* wtyp = attn + BB * LL * LL;      // B*L*64 (padded 47->64)
    float* agg  = wtyp + BB * LL * 64;      // B*L*D
    float* pool = agg + BB * LL * DD;       // NL*B*D
    float* out  = (float*)d_out;            // B*R = 512 floats

    k_gram<<<9, 256, 0, stream>>>(table, G);
    k_bn<<<(BB * LL * DD) / 256, 256, 0, stream>>>(text, gamma, beta, seq0);

    float* sin  = seq0;
    float* sout = seq1;
    for (int l = 0; l < NL; ++l) {
        k_score<<<72, 128, 0, stream>>>(sin, G, depv, attn);          // 288 waves
        k_attn<<<BB * LL, 128, 0, stream>>>(attn, adj, depv, attn, wtyp);
        k_agg<<<192, 256, 0, stream>>>(attn, sin, wtyp, table, agg);  // 1536 waves
        k_fc<<<192, 256, 0, stream>>>(agg, Wst + (size_t)l * DD * DD,
                                      bst + (size_t)l * DD, sout);    // 1536 waves
        k_pool<<<BB, DD, 0, stream>>>(sout, mask, pool + (size_t)l * BB * DD);
        float* t = sin; sin = sout; sout = t;
    }
    k_out<<<1, BB * RR, 0, stream>>>(pool, enslin, fcW, fcb, out);
}